// LossLayer_80831284511303
// MI455X (gfx1250) — compile-verified
//
#include <hip/hip_runtime.h>
#include <hip/hip_bf16.h>
#include <stdint.h>

// ---------------- problem constants (from reference) ----------------
#define GSZ    52
#define NANCH  3
#define NCELL  (GSZ * GSZ * NANCH)   // 8112
#define NBATCH 32
#define NCLS   80
#define CSTRIDE 85                   // 5 + 80 floats per anchor-cell
#define MAXGT  64
#define RATIO  8.0f                  // 416/52, exact power of two
#define IMGSZ  416.0f

typedef unsigned int u32x4 __attribute__((ext_vector_type(4)));
typedef int          i32x8 __attribute__((ext_vector_type(8)));
typedef int          i32x4 __attribute__((ext_vector_type(4)));

__device__ __forceinline__ float softplusf(float x) {
    // jax.nn.softplus = max(x,0) + log1p(exp(-|x|)); 1+e is in [1,2] so plain log is fine
    return fmaxf(x, 0.f) + __logf(1.f + __expf(-fabsf(x)));
}
__device__ __forceinline__ float sigmoidf(float x) {
    return 1.f / (1.f + __expf(-x));
}
__device__ __forceinline__ float logclipf(float v) {
    return __logf(fminf(fmaxf(v, 1e-9f), 1e9f));
}

// =====================================================================
// Kernel 1: deterministic compaction of GT (conf>0) cells, per batch.
// One 256-thread block per batch; block-wide exclusive scan gives a
// schedule-independent slot assignment (no atomics -> bitwise repeatable).
// =====================================================================
__global__ void k_gather(const float* __restrict__ tgt,
                         float4* __restrict__ gtbox,
                         int*    __restrict__ gtidx,
                         int*    __restrict__ gtcnt)
{
    const int b   = blockIdx.x;
    const int tid = threadIdx.x;
    __shared__ int scan[256];
    const float* tb = tgt + (size_t)b * NCELL * CSTRIDE;

    int cnt = 0;
    for (int it = 0; it < 32; ++it) {
        int n = it * 256 + tid;
        if (n < NCELL && tb[(size_t)n * CSTRIDE + 4] > 0.f) ++cnt;
    }
    scan[tid] = cnt;
    __syncthreads();
    for (int d = 1; d < 256; d <<= 1) {           // Hillis-Steele inclusive scan
        int v = (tid >= d) ? scan[tid - d] : 0;
        __syncthreads();
        scan[tid] += v;
        __syncthreads();
    }
    int ofs = scan[tid] - cnt;                    // exclusive prefix
    for (int it = 0; it < 32; ++it) {
        int n = it * 256 + tid;
        if (n < NCELL) {
            const float* c = tb + (size_t)n * CSTRIDE;
            if (c[4] > 0.f) {
                if (ofs < MAXGT) {
                    gtbox[b * MAXGT + ofs] = make_float4(c[0], c[1], c[2], c[3]);
                    gtidx[b * MAXGT + ofs] = n;
                }
                ++ofs;
            }
        }
    }
    if (tid == 255) gtcnt[b] = min(scan[255], MAXGT);
}

// =====================================================================
// Kernel 2: noobj confidence loss over ALL cells.
// GT boxes (1 KB/batch) are broadcast into LDS via the Tensor Data Mover
// (tensor_load_to_lds + s_wait_tensorcnt). One thread per anchor-cell.
// Reads only P[0..4] and T[4]  ->  ~6 floats/cell instead of 170.
// =====================================================================
__global__ void k_noobj(const float* __restrict__ preds,
                        const float* __restrict__ tgt,
                        const float* __restrict__ anch,
                        const float4* __restrict__ gtbox,
                        const int*   __restrict__ gtcnt,
                        float* __restrict__ partials)
{
    const int b    = blockIdx.y;
    const int tid  = threadIdx.x;
    const int cell = blockIdx.x * 256 + tid;
    __shared__ float4 sGT[MAXGT];
    __shared__ float  sRed[256];

    const float* P = preds + ((size_t)b * NCELL + (cell < NCELL ? cell : NCELL - 1)) * CSTRIDE;
    __builtin_prefetch(P, 0, 1);   // global_prefetch: warm the cell header while TDM runs

    // ---- TDM: DMA 64 float4 GT boxes (256 dwords) from L2/ws into LDS ----
    if (tid < 32) {
        const uint64_t ga  = (uint64_t)(uintptr_t)(gtbox + (size_t)b * MAXGT);
        const uint32_t lds = (uint32_t)(uintptr_t)&sGT[0];   // low 32 bits = LDS byte offset
        u32x4 g0; i32x8 g1; i32x4 g2, g3; i32x8 g4;
        // D# group0: count=1 | lds_addr | global_addr[56:0] | type=2
        g0[0] = 1u;
        g0[1] = lds;
        g0[2] = (uint32_t)ga;
        g0[3] = (uint32_t)((ga >> 32) & 0x1FFFFFFu) | (2u << 30);
        // D# group1: data_size=2 (4B), tensor_dim0=256, tensor_dim1=1,
        //            tile_dim0=256 (1-D tile), tensor_dim0_stride=256
        g1[0] = (int)(2u << 16);
        g1[1] = (int)(256u << 16);      // tensor_dim0[15:0] @ bits 63:48
        g1[2] = (int)(1u << 16);        // tensor_dim0[31:16]=0 | tensor_dim1[15:0]=1
        g1[3] = (int)(256u << 16);      // tensor_dim1[31:16]=0 | tile_dim0=256
        g1[4] = 0;                      // tile_dim1=0, tile_dim2=0 (unused)
        g1[5] = 256;                    // tensor_dim0_stride
        g1[6] = 0; g1[7] = 0;
        g2[0] = 1; g2[1] = 1; g2[2] = 0; g2[3] = 0;         // tensor_dim2=1, tensor_dim3=1
        g3[0] = 0; g3[1] = (int)(1u << 16); g3[2] = 0; g3[3] = 0;  // tensor_dim4=1
        g4[0] = 0; g4[1] = 0; g4[2] = 0; g4[3] = 0;         // extra group (clang-23 form):
        g4[4] = 0; g4[5] = 0; g4[6] = 0; g4[7] = 0;         // zero-filled (probe-verified OK)
        __builtin_amdgcn_tensor_load_to_lds(g0, g1, g2, g3, g4, 0);
        __builtin_amdgcn_s_wait_tensorcnt(0);
    }
    __syncthreads();

    float part = 0.f;
    if (cell < NCELL) {
        const int   a    = cell % NANCH;
        const int   gl   = cell / NANCH;
        const float offx = (float)(gl % GSZ);    // meshgrid 'xy': x = column
        const float offy = (float)(gl / GSZ);
        const float xr = P[0], yr = P[1], wr = P[2], hr = P[3], cl = P[4];
        const float tconf = tgt[((size_t)b * NCELL + cell) * CSTRIDE + 4];
        const float aw = anch[a * 2 + 0], ah = anch[a * 2 + 1];

        const float sx = sigmoidf(xr), sy = sigmoidf(yr);
        const float bx = (sx + offx) * RATIO, by = (sy + offy) * RATIO;
        const float bw = __expf(wr) * aw,     bh = __expf(hr) * ah;

        const float obj = (tconf > 0.f) ? 1.f : 0.f;
        const float bce = softplusf(cl) - cl * tconf;

        const float px0 = bx - bw * 0.5f, px1 = bx + bw * 0.5f;
        const float py0 = by - bh * 0.5f, py1 = by + bh * 0.5f;
        const float areaP = bw * bh;
        const int   cnt = gtcnt[b];
        float best = 0.f;
        for (int k = 0; k < cnt; ++k) {
            const float4 g = sGT[k];
            const float gx0 = g.x - g.z * 0.5f, gx1 = g.x + g.z * 0.5f;
            const float gy0 = g.y - g.w * 0.5f, gy1 = g.y + g.w * 0.5f;
            const float iw = fmaxf(fminf(px1, gx1) - fmaxf(px0, gx0), 0.f);
            const float ih = fmaxf(fminf(py1, gy1) - fmaxf(py0, gy0), 0.f);
            const float inter = iw * ih;
            const float uni   = (areaP + g.z * g.w) - inter + 1e-9f;
            best = fmaxf(best, inter * __builtin_amdgcn_rcpf(uni));
        }
        const float noobj = (best < 0.6f) ? 1.f : 0.f;
        part = (1.f - obj) * noobj * bce;
    }

    // fixed-order block tree reduction (deterministic)
    sRed[tid] = part;
    __syncthreads();
    for (int d = 128; d >= 1; d >>= 1) {
        if (tid < d) sRed[tid] += sRed[tid + d];
        __syncthreads();
    }
    if (tid == 0) partials[blockIdx.y * 32 + blockIdx.x] = sRed[0];
}

// =====================================================================
// Kernel 3: object-cell losses (txty, twth, obj-conf, class BCE).
// One wave32 per gathered GT entry; lanes split the 80 classes and
// butterfly-reduce with __shfl_xor (ds-permute path).
// =====================================================================
__global__ void k_obj(const float* __restrict__ preds,
                      const float* __restrict__ tgt,
                      const float* __restrict__ anch,
                      const int*   __restrict__ gtidx,
                      const int*   __restrict__ gtcnt,
                      float* __restrict__ partials)     // 4 floats per block
{
    const int tid  = threadIdx.x;
    const int w    = tid >> 5, lane = tid & 31;
    const int e    = blockIdx.x * 8 + w;                // entry id in [0, B*64)
    const int b    = e >> 6, slot = e & 63;
    __shared__ float sRed[8][4];

    float v0 = 0.f, v1 = 0.f, v2 = 0.f, v3 = 0.f;
    if (slot < gtcnt[b]) {
        const int    n = gtidx[b * MAXGT + slot];
        const float* P = preds + ((size_t)b * NCELL + n) * CSTRIDE;
        const float* T = tgt   + ((size_t)b * NCELL + n) * CSTRIDE;

        float cls = 0.f;                                // 80 classes over 32 lanes
        for (int c = lane; c < NCLS; c += 32) {
            const float x = P[5 + c];
            cls += softplusf(x) - x * T[5 + c];
        }
        for (int m = 16; m >= 1; m >>= 1) cls += __shfl_xor(cls, m, 32);

        const int   a    = n % NANCH, gl = n / NANCH;
        const float offx = (float)(gl % GSZ), offy = (float)(gl / GSZ);
        const float aw = anch[a * 2], ah = anch[a * 2 + 1];
        const float xr = P[0], yr = P[1], wr = P[2], hr = P[3], cl2 = P[4];
        const float tx = T[0], ty = T[1], tw = T[2], th = T[3], tconf = T[4];

        // replicate reference fp ordering: box_xy/ratio - offset == (sig+off)-off (ratio=8 exact)
        const float sx = sigmoidf(xr), sy = sigmoidf(yr);
        const float pxx = (sx + offx) - offx, pxy = (sy + offy) - offy;
        const float txx = tx * (1.f / RATIO) - offx, txy = ty * (1.f / RATIO) - offy;

        const float bw = __expf(wr) * aw, bh = __expf(hr) * ah;
        float pwx = bw / aw, pwy = bh / ah;
        float twx = tw / aw, twy = th / ah;
        twx = (twx == 0.f) ? 1.f : twx;  twy = (twy == 0.f) ? 1.f : twy;
        pwx = (pwx == 0.f) ? 1.f : pwx;  pwy = (pwy == 0.f) ? 1.f : pwy;
        const float ltx = logclipf(twx), lty = logclipf(twy);
        const float lpx = logclipf(pwx), lpy = logclipf(pwy);

        const float scale = 2.f - ((tw / IMGSZ) * th) / IMGSZ;
        const float dx = pxx - txx, dy = pxy - txy;
        const float dwx = lpx - ltx, dwy = lpy - lty;

        v0 = (dx * dx + dy * dy) * scale;     // obj == 1 for gathered cells
        v1 = (dwx * dwx + dwy * dwy) * scale;
        v2 = softplusf(cl2) - cl2 * tconf;
        v3 = cls;
    }
    if (lane == 0) { sRed[w][0] = v0; sRed[w][1] = v1; sRed[w][2] = v2; sRed[w][3] = v3; }
    __syncthreads();
    if (tid < 4) {                                     // fixed-order 8-wave sum
        float s = 0.f;
        for (int i = 0; i < 8; ++i) s += sRed[i][tid];
        partials[blockIdx.x * 4 + tid] = s;
    }
}

// =====================================================================
// Kernel 4: final fixed-order reduction, /B, total.  Output order:
// (txty, twth, noobj_conf, obj_conf, class, total)
// =====================================================================
__global__ void k_final(const float* __restrict__ pB,   // 1024 noobj partials
                        const float* __restrict__ pC,   // 256 x 4 obj partials
                        float* __restrict__ out)
{
    __shared__ float red[256];
    const int tid = threadIdx.x;

    float s = 0.f;
    for (int i = tid; i < 1024; i += 256) s += pB[i];
    red[tid] = s;
    __syncthreads();
    for (int d = 128; d >= 1; d >>= 1) { if (tid < d) red[tid] += red[tid + d]; __syncthreads(); }
    const float noobj_sum = red[0];
    __syncthreads();

    float sums[4];
    for (int j = 0; j < 4; ++j) {
        red[tid] = pC[tid * 4 + j];
        __syncthreads();
        for (int d = 128; d >= 1; d >>= 1) { if (tid < d) red[tid] += red[tid + d]; __syncthreads(); }
        sums[j] = red[0];
        __syncthreads();
    }
    if (tid == 0) {
        const float invB = 1.f / (float)NBATCH;
        const float txty = sums[0] * invB;
        const float twth = sums[1] * invB;
        const float noc  = noobj_sum * invB;
        const float oc   = sums[2] * invB;
        const float cls  = sums[3] * invB;
        out[0] = txty; out[1] = twth; out[2] = noc; out[3] = oc; out[4] = cls;
        out[5] = txty + twth + noc + oc + cls;
    }
}

// =====================================================================
extern "C" void kernel_launch(void* const* d_in, const int* in_sizes, int n_in,
                              void* d_out, int out_size, void* d_ws, size_t ws_size,
                              hipStream_t stream)
{
    const float* preds = (const float*)d_in[0];   // (32,52,52,255) f32
    const float* tgt   = (const float*)d_in[1];   // (32,52,52,3,85) f32
    const float* anch  = (const float*)d_in[2];   // (3,2) f32

    char*   ws    = (char*)d_ws;
    float4* gtbox = (float4*)(ws);                          // 32*64*16 = 32768 B
    int*    gtidx = (int*)(ws + 32768);                     //  8192 B
    int*    gtcnt = (int*)(ws + 32768 + 8192);              //   128 B
    float*  pB    = (float*)(ws + 32768 + 8192 + 128);      //  4096 B (1024 f32)
    float*  pC    = (float*)(ws + 32768 + 8192 + 128 + 4096); // 4096 B (256*4 f32)

    k_gather<<<NBATCH, 256, 0, stream>>>(tgt, gtbox, gtidx, gtcnt);
    k_noobj <<<dim3(32, NBATCH), 256, 0, stream>>>(preds, tgt, anch, gtbox, gtcnt, pB);
    k_obj   <<<256, 256, 0, stream>>>(preds, tgt, anch, gtidx, gtcnt, pC);
    k_final <<<1, 256, 0, stream>>>(pB, pC, (float*)d_out);
}